// DeVRF_39745627357198
// MI455X (gfx1250) — compile-verified
//
#include <hip/hip_runtime.h>
#include <math.h>

// ---------------------------------------------------------------------------
// DeVRF volume rendering for MI455X (gfx1250, wave32).
//   k1: per-sample trilerp(density)->softplus->s ; trilerp(k0)->sigmoid->rgb
//       + sorted-segment boundary detection (no atomics).
//   k2: one wave per ray; wave-level scan for transmittance; WMMA f32 16x16x4
//       (all-ones B => layout-safe) for the final 32-lane RGB reduction.
// ---------------------------------------------------------------------------

typedef __attribute__((ext_vector_type(2))) float v2f;
typedef __attribute__((ext_vector_type(8))) float v8f;

#define ACT_SHIFT  (-13.815509557935518f)   // log(1/(1-1e-6) - 1)
#define GRID_RES   256
#define STRIDE_X   65536                    // 256*256
#define STRIDE_Y   256
#define CH_STRIDE  16777216                 // 256^3 (k0 channel stride)

// ---- 32-lane sum via V_WMMA_F32_16X16X4_F32 -------------------------------
// A (16x4, documented layout): lane m (0..15): {A[m][0], A[m][1]} ; lane 16+m:
// {A[m][2], A[m][3]}.  Put v in A.x, 0 in A.y.  B = all ones (value uniform =>
// any B layout gives the same product).  D[m][n] = v[m] + v[m+16].
// D layout (documented): lane L<16 holds D[0..7][L], lane L>=16 holds
// D[8..15][L-16].  sum of 8 D regs + shfl_xor(16) = full 32-lane sum.
__device__ __forceinline__ float wave_sum_wmma(float v) {
    v2f a; a.x = v;    a.y = 0.0f;
    v2f b; b.x = 1.0f; b.y = 1.0f;
    v8f c = {};
    v8f d = __builtin_amdgcn_wmma_f32_16x16x4_f32(
        /*neg_a=*/false, a, /*neg_b=*/false, b,
        /*c_mod=*/(short)0, c, /*reuse_a=*/false, /*reuse_b=*/false);
    float r = d[0] + d[1] + d[2] + d[3] + d[4] + d[5] + d[6] + d[7];
    r += __shfl_xor(r, 16, 32);
    return r;
}

__device__ __forceinline__ void grid_coord(float p, int& i0, float& f) {
    float ind = (p + 1.0f) * 127.5f;                 // (p-XYZ_MIN)/range*(res-1)
    ind = fminf(fmaxf(ind, 0.0f), 255.0f);
    int ii = (int)ind;                               // floor (non-negative)
    ii = min(ii, GRID_RES - 2);
    i0 = ii;
    f  = ind - (float)ii;
}

__device__ __forceinline__ float trilerp_fetch(const float* __restrict__ g,
                                               int base, const float* w) {
    float acc = 0.0f;
    acc += w[0] * g[base];
    acc += w[1] * g[base + 1];
    acc += w[2] * g[base + STRIDE_Y];
    acc += w[3] * g[base + STRIDE_Y + 1];
    acc += w[4] * g[base + STRIDE_X];
    acc += w[5] * g[base + STRIDE_X + 1];
    acc += w[6] * g[base + STRIDE_X + STRIDE_Y];
    acc += w[7] * g[base + STRIDE_X + STRIDE_Y + 1];
    return acc;
}

__global__ void init_rays(int* __restrict__ rstart, int* __restrict__ rend, int N) {
    int r = blockIdx.x * blockDim.x + threadIdx.x;
    if (r < N) { rstart[r] = 0; rend[r] = -1; }      // empty ray => loop skipped
}

__global__ void sample_kernel(const float* __restrict__ dens,
                              const float* __restrict__ k0,
                              const float* __restrict__ xyz,
                              const int*   __restrict__ ray_id,
                              float* __restrict__ s_out,
                              float* __restrict__ rgb_out,
                              int*   __restrict__ rstart,
                              int*   __restrict__ rend,
                              int M) {
    int i = blockIdx.x * blockDim.x + threadIdx.x;
    if (i >= M) return;

    float px = xyz[3 * i + 0];
    float py = xyz[3 * i + 1];
    float pz = xyz[3 * i + 2];

    int x0, y0, z0; float fx, fy, fz;
    grid_coord(px, x0, fx);
    grid_coord(py, y0, fy);
    grid_coord(pz, z0, fz);

    float gx = 1.0f - fx, gy = 1.0f - fy, gz = 1.0f - fz;
    float w[8];
    w[0] = gx * gy * gz;  w[1] = gx * gy * fz;
    w[2] = gx * fy * gz;  w[3] = gx * fy * fz;
    w[4] = fx * gy * gz;  w[5] = fx * gy * fz;
    w[6] = fx * fy * gz;  w[7] = fx * fy * fz;

    int base = x0 * STRIDE_X + y0 * STRIDE_Y + z0;

    // density -> softplus (numerically stable) ; INTERVAL == 1
    float raw = trilerp_fetch(dens, base, w) + ACT_SHIFT;
    float s   = fmaxf(raw, 0.0f) + log1pf(expf(-fabsf(raw)));
    s_out[i]  = s;

    // k0 -> sigmoid, 3 channels
    #pragma unroll
    for (int c = 0; c < 3; ++c) {
        float v = trilerp_fetch(k0 + (size_t)c * CH_STRIDE, base, w);
        rgb_out[3 * i + c] = 1.0f / (1.0f + expf(-v));
    }

    // sorted-segment boundaries (no atomics needed: ray_id is sorted)
    int rid = ray_id[i];
    if (i == 0 || ray_id[i - 1] != rid) rstart[rid] = i;
    if (i == M - 1 || ray_id[i + 1] != rid) rend[rid] = i;
}

__global__ void ray_march_kernel(const float* __restrict__ s_arr,
                                 const float* __restrict__ rgb,
                                 const int*   __restrict__ rstart,
                                 const int*   __restrict__ rend,
                                 float* __restrict__ out,
                                 int N) {
    int wave = (blockIdx.x * blockDim.x + threadIdx.x) >> 5;  // one wave per ray
    int lane = threadIdx.x & 31;
    if (wave >= N) return;                                    // wave-uniform

    int start = rstart[wave];
    int end   = rend[wave];                                   // inclusive

    float cum = 0.0f;                 // within-ray prefix of s (wave-uniform)
    float ar = 0.0f, ag = 0.0f, ab = 0.0f;

    for (int base = start; base <= end; base += 32) {
        int  i     = base + lane;
        bool valid = (i <= end);
        int  ic    = valid ? i : end;                         // safe index
        float s    = s_arr[ic];
        s = valid ? s : 0.0f;                                 // alpha=0, scan no-op

        // wave32 inclusive scan of s
        float scan = s;
        #pragma unroll
        for (int off = 1; off < 32; off <<= 1) {
            float n = __shfl_up(scan, off, 32);
            if (lane >= off) scan += n;
        }

        float T     = expf(-(cum + (scan - s)));              // exclusive prefix
        float alpha = 1.0f - expf(-s);
        float wgt   = alpha * T;                              // 0 for invalid lanes

        ar += wgt * rgb[3 * ic + 0];
        ag += wgt * rgb[3 * ic + 1];
        ab += wgt * rgb[3 * ic + 2];

        cum += __shfl(scan, 31, 32);                          // chunk total
    }

    // EXEC is all-ones here (uniform control flow, exact wave count)
    float sr = wave_sum_wmma(ar);
    float sg = wave_sum_wmma(ag);
    float sb = wave_sum_wmma(ab);

    if (lane == 0) {
        float last = expf(-cum);                              // alphainv_last
        out[3 * wave + 0] = sr + last;
        out[3 * wave + 1] = sg + last;
        out[3 * wave + 2] = sb + last;
    }
}

extern "C" void kernel_launch(void* const* d_in, const int* in_sizes, int n_in,
                              void* d_out, int out_size, void* d_ws, size_t ws_size,
                              hipStream_t stream) {
    const float* dens   = (const float*)d_in[0];   // [256,256,256]
    const float* k0     = (const float*)d_in[1];   // [3,256,256,256]
    const float* xyz    = (const float*)d_in[2];   // [M,3]
    const int*   ray_id = (const int*)d_in[3];     // [M]
    (void)d_in; (void)n_in; (void)ws_size;

    const int M = in_sizes[3];
    const int N = out_size / 3;                    // rgb_marched is [N,3]

    float* s_buf   = (float*)d_ws;                 // M floats
    float* rgb_buf = s_buf + (size_t)M;            // 3M floats
    int*   r_start = (int*)(rgb_buf + (size_t)3 * M);
    int*   r_end   = r_start + N;
    float* out     = (float*)d_out;

    init_rays<<<(N + 255) / 256, 256, 0, stream>>>(r_start, r_end, N);
    sample_kernel<<<(M + 255) / 256, 256, 0, stream>>>(
        dens, k0, xyz, ray_id, s_buf, rgb_buf, r_start, r_end, M);
    ray_march_kernel<<<((size_t)N * 32 + 255) / 256, 256, 0, stream>>>(
        s_buf, rgb_buf, r_start, r_end, out, N);
}